// RMIHieraTripletLoss_47991964565836
// MI455X (gfx1250) — compile-verified
//
#include <hip/hip_runtime.h>
#include <hip/hip_bf16.h>
#include <math.h>

// ---------------------------------------------------------------------------
// RMI Hiera Triplet Loss forward, MI455X (gfx1250, wave32).
// cls_score (121.6MB) is read exactly once (fused pool+BCE, ~5.2us @23.3TB/s),
// pooled arrays (~27MB) live in workspace / L2, Gram matrices computed with
// V_WMMA_F32_16X16X4_F32 split across 4 blocks per (b,c) slot for occupancy.
// ---------------------------------------------------------------------------

typedef float v2f __attribute__((ext_vector_type(2)));
typedef float v8f __attribute__((ext_vector_type(8)));

#define NF 19
#define NM 7
#define NH 3
#define NC 29
#define HH 512
#define WW 512
#define HWB (HH * WW)          // 262144
#define BB 4
#define NPOOL 170              // floor((512-3)/3)+1
#define PP (NPOOL * NPOOL)     // 28900
#define NWIN 168               // NPOOL - 3 + 1
#define NPAIR (NWIN * NWIN)    // 28224  (divisible by 4)
#define CLIPV 1e-6f
#define ALPHA 1e-3f
#define GSPLIT 4               // blocks per (b,c) gram slot

constexpr int F2M[NF] = {0,0,0,1,1,1,2,2,3,3,4,5,5,6,6,6,6,6,6};
constexpr int M2H[NM] = {0,0,1,1,1,2,2};

// ---- workspace layout (floats) --------------------------------------------
constexpr long long OFF_LAF = 0;
constexpr long long SZ_F    = (long long)BB * NF * PP;   // 2,196,400
constexpr long long OFF_PRF = OFF_LAF + SZ_F;
constexpr long long OFF_LAM = OFF_PRF + SZ_F;
constexpr long long SZ_M    = (long long)BB * NM * PP;   // 809,200
constexpr long long OFF_PRM = OFF_LAM + SZ_M;
constexpr long long OFF_LAH = OFF_PRM + SZ_M;
constexpr long long SZ_H    = (long long)BB * NH * PP;   // 346,800
constexpr long long OFF_PRH = OFF_LAH + SZ_H;
constexpr long long OFF_GRAM = OFF_PRH + SZ_H;
constexpr long long N_SLOTS  = (long long)BB * (NF + NM + NH); // 116
constexpr long long GRAM_STRIDE = 272; // 3*81 grams + 9+9 sums, padded
constexpr long long OFF_SCAL = OFF_GRAM + N_SLOTS * GRAM_STRIDE;
// scal: [0..2] bce sums, [3] valid count, [4..6] rmi logdet sums

// ---------------------------------------------------------------------------
__device__ __forceinline__ void load_probs(const float* __restrict__ p,
                                           float* pf, float* pm, float* ph) {
#pragma unroll
  for (int f = 0; f < NF; ++f) pf[f] = 1.f / (1.f + __expf(-p[(long long)f * HWB]));
#pragma unroll
  for (int m = 0; m < NM; ++m) pm[m] = 1.f / (1.f + __expf(-p[(long long)(NF + m) * HWB]));
#pragma unroll
  for (int k = 0; k < NH; ++k) ph[k] = 1.f / (1.f + __expf(-p[(long long)(NF + NM + k) * HWB]));
}

// BCE for one pixel. Restructured so only one log(pos) per level is taken:
//   sum_c BCE_c = sum_c log(1-neg_c) - log(1-neg_lab) + log(pos_lab)
// selections done with unrolled compare-selects (stays in VGPRs).
__device__ __forceinline__ void bce_pixel(const float* pf, const float* pm, const float* ph,
                                          int lf, int lm, int lh, float vm,
                                          float& s0, float& s1, float& s2) {
  float segf[NM], mbc[NM], segm[NH], mcc[NH];
#pragma unroll
  for (int m = 0; m < NM; ++m) segf[m] = -1e30f;
#pragma unroll
  for (int f = 0; f < NF; ++f) segf[F2M[f]] = fmaxf(segf[F2M[f]], pf[f]);
#pragma unroll
  for (int m = 0; m < NM; ++m) mbc[m] = fmaxf(segf[m], pm[m]);   // MCMB_c
#pragma unroll
  for (int k = 0; k < NH; ++k) segm[k] = -1e30f;
#pragma unroll
  for (int m = 0; m < NM; ++m) segm[M2H[m]] = fmaxf(segm[M2H[m]], mbc[m]);
#pragma unroll
  for (int k = 0; k < NH; ++k) mcc[k] = fmaxf(segm[k], ph[k]);   // MCMC_c

  { // fine: pos = MCLA = min(pf, pm[F2M]), neg = MCMA = pf
    float acc = 0.f, selPos = 1.f, selL1 = 0.f;
#pragma unroll
    for (int f = 0; f < NF; ++f) {
      float l1 = __logf(1.f - pf[f] + 1e-8f);
      acc += l1;
      bool hit = (f == lf);
      selL1 = hit ? l1 : selL1;
      float pos = fminf(pf[f], pm[F2M[f]]);
      selPos = hit ? pos : selPos;
    }
    s0 -= vm * (acc - selL1 + __logf(selPos + 1e-8f));
  }
  { // mid: pos = MCLB_c = min(pm, ph[M2H]), neg = MCMB_c = mbc
    float acc = 0.f, selPos = 1.f, selL1 = 0.f;
#pragma unroll
    for (int m = 0; m < NM; ++m) {
      float l1 = __logf(1.f - mbc[m] + 1e-8f);
      acc += l1;
      bool hit = (m == lm);
      selL1 = hit ? l1 : selL1;
      float pos = fminf(pm[m], ph[M2H[m]]);
      selPos = hit ? pos : selPos;
    }
    s1 -= vm * (acc - selL1 + __logf(selPos + 1e-8f));
  }
  { // high: pos = MCLC_c = ph, neg = MCMC_c = mcc
    float acc = 0.f, selPos = 1.f, selL1 = 0.f;
#pragma unroll
    for (int k = 0; k < NH; ++k) {
      float l1 = __logf(1.f - mcc[k] + 1e-8f);
      acc += l1;
      bool hit = (k == lh);
      selL1 = hit ? l1 : selL1;
      selPos = hit ? ph[k] : selPos;
    }
    s2 -= vm * (acc - selL1 + __logf(selPos + 1e-8f));
  }
}

__device__ __forceinline__ void reduce4_atomic(float s0, float s1, float s2, float sv,
                                               float* __restrict__ scal) {
#pragma unroll
  for (int off = 16; off > 0; off >>= 1) {
    s0 += __shfl_down(s0, off, 32);
    s1 += __shfl_down(s1, off, 32);
    s2 += __shfl_down(s2, off, 32);
    sv += __shfl_down(sv, off, 32);
  }
  __shared__ float sm[4][8];
  int lane = threadIdx.x & 31, wv = threadIdx.x >> 5;
  if (lane == 0) { sm[0][wv] = s0; sm[1][wv] = s1; sm[2][wv] = s2; sm[3][wv] = sv; }
  __syncthreads();
  if (threadIdx.x == 0) {
    float t0 = 0, t1 = 0, t2 = 0, t3 = 0;
    int nw = (blockDim.x + 31) >> 5;
    for (int w = 0; w < nw; ++w) { t0 += sm[0][w]; t1 += sm[1][w]; t2 += sm[2][w]; t3 += sm[3][w]; }
    atomicAdd(&scal[0], t0); atomicAdd(&scal[1], t1);
    atomicAdd(&scal[2], t2); atomicAdd(&scal[3], t3);
  }
}

// ---------------------------------------------------------------------------
// Kernel 1: fused 3x3/stride-3 maxpool of la/pr (all 3 levels) + BCE over the
// 510x510 pooled region. One thread per pooled cell (owns 9 distinct pixels).
// ---------------------------------------------------------------------------
__global__ void pool_bce_kernel(const float* __restrict__ cls, const int* __restrict__ label,
                                float* __restrict__ ws) {
  float* laF = ws + OFF_LAF; float* prF = ws + OFF_PRF;
  float* laM = ws + OFF_LAM; float* prM = ws + OFF_PRM;
  float* laH = ws + OFF_LAH; float* prH = ws + OFF_PRH;
  float* scal = ws + OFF_SCAL;

  int idx = blockIdx.x * blockDim.x + threadIdx.x;
  float s0 = 0.f, s1 = 0.f, s2 = 0.f, sv = 0.f;
  if (idx < BB * PP) {
    int b = idx / PP;
    int r = idx - b * PP;
    int py = r / NPOOL, px = r - py * NPOOL;

    float laf[NF], prf[NF], lam[NM], prm[NM], lah[NH], prh[NH];
#pragma unroll
    for (int f = 0; f < NF; ++f) { laf[f] = -1e30f; prf[f] = -1e30f; }
#pragma unroll
    for (int m = 0; m < NM; ++m) { lam[m] = -1e30f; prm[m] = -1e30f; }
#pragma unroll
    for (int k = 0; k < NH; ++k) { lah[k] = -1e30f; prh[k] = -1e30f; }

#pragma unroll
    for (int iy = 0; iy < 3; ++iy)
#pragma unroll
      for (int ix = 0; ix < 3; ++ix) {
        int hh = py * 3 + iy, wc = px * 3 + ix;
        int lab = label[(long long)b * HWB + hh * WW + wc];
        bool valid = (lab != 255);
        int lf = valid ? lab : 0;
        lf = (lf < 0) ? 0 : ((lf >= NF) ? NF - 1 : lf);
        float vm = valid ? 1.f : 0.f;
        const float* p = cls + (long long)b * NC * HWB + hh * WW + wc;
        float pf[NF], pm[NM], ph[NH];
        load_probs(p, pf, pm, ph);
        int lm = F2M[lf], lh = M2H[lm];
        bce_pixel(pf, pm, ph, lf, lm, lh, vm, s0, s1, s2);
        sv += vm;
#pragma unroll
        for (int f = 0; f < NF; ++f) {
          laf[f] = fmaxf(laf[f], (f == lf) ? vm : 0.f);
          prf[f] = fmaxf(prf[f], fminf(pf[f], pm[F2M[f]]) * vm + CLIPV);
        }
#pragma unroll
        for (int m = 0; m < NM; ++m) {
          lam[m] = fmaxf(lam[m], (m == lm) ? vm : 0.f);
          prm[m] = fmaxf(prm[m], fminf(pm[m], ph[M2H[m]]) * vm + CLIPV);
        }
#pragma unroll
        for (int k = 0; k < NH; ++k) {
          lah[k] = fmaxf(lah[k], (k == lh) ? vm : 0.f);
          prh[k] = fmaxf(prh[k], ph[k] * vm + CLIPV);
        }
      }
#pragma unroll
    for (int f = 0; f < NF; ++f) {
      long long o = (((long long)b * NF + f) * NPOOL + py) * NPOOL + px;
      laF[o] = laf[f]; prF[o] = prf[f];
    }
#pragma unroll
    for (int m = 0; m < NM; ++m) {
      long long o = (((long long)b * NM + m) * NPOOL + py) * NPOOL + px;
      laM[o] = lam[m]; prM[o] = prm[m];
    }
#pragma unroll
    for (int k = 0; k < NH; ++k) {
      long long o = (((long long)b * NH + k) * NPOOL + py) * NPOOL + px;
      laH[o] = lah[k]; prH[o] = prh[k];
    }
  }
  reduce4_atomic(s0, s1, s2, sv, scal);
}

// ---------------------------------------------------------------------------
// Kernel 2: BCE on the fringe pixels (rows/cols 510,511) not covered above.
// 512^2 - 510^2 = 2044 pixels per batch.
// ---------------------------------------------------------------------------
__global__ void border_bce_kernel(const float* __restrict__ cls, const int* __restrict__ label,
                                  float* __restrict__ scal) {
  const int PERB = HWB - 510 * 510; // 2044
  int idx = blockIdx.x * blockDim.x + threadIdx.x;
  float s0 = 0.f, s1 = 0.f, s2 = 0.f, sv = 0.f;
  if (idx < BB * PERB) {
    int b = idx / PERB, r = idx - b * PERB;
    int hh, wc;
    if (r < 2 * WW) { hh = 510 + (r >> 9); wc = r & 511; }          // rows 510,511
    else { int j = r - 2 * WW; hh = j >> 1; wc = 510 + (j & 1); }   // cols 510,511
    int lab = label[(long long)b * HWB + hh * WW + wc];
    bool valid = (lab != 255);
    int lf = valid ? lab : 0;
    lf = (lf < 0) ? 0 : ((lf >= NF) ? NF - 1 : lf);
    float vm = valid ? 1.f : 0.f;
    const float* p = cls + (long long)b * NC * HWB + hh * WW + wc;
    float pf[NF], pm[NM], ph[NH];
    load_probs(p, pf, pm, ph);
    bce_pixel(pf, pm, ph, lf, F2M[lf], M2H[F2M[lf]], vm, s0, s1, s2);
    sv += vm;
  }
  reduce4_atomic(s0, s1, s2, sv, scal);
}

// ---------------------------------------------------------------------------
// Kernel 3: Gram matrices via V_WMMA_F32_16X16X4_F32.
// Per (b,c): X = la_v (9 x 28224), Y = pr_v. Computes raw X·Xᵀ, Y·Yᵀ, X·Yᵀ
// plus row sums. d=0..8 → M/N rows (padded to 16 with a lane mask), 4 pair
// indices per WMMA step along K. A(16x4) and B(4x16) register images coincide
// per the ISA layouts, so one v2f load per lane feeds both operands.
// GSPLIT blocks cooperate on one (b,c) slot (atomic merge) for occupancy.
// ---------------------------------------------------------------------------
__global__ void gram_kernel(const float* __restrict__ la_pool, const float* __restrict__ pr_pool,
                            float* __restrict__ gram) {
  int bc   = blockIdx.x / GSPLIT;
  int part = blockIdx.x - bc * GSPLIT;
  const float* la = la_pool + (long long)bc * PP;
  const float* pr = pr_pool + (long long)bc * PP;
  float* g = gram + (long long)bc * GRAM_STRIDE;

  int lane = threadIdx.x & 31;
  int wave = threadIdx.x >> 5;
  int M = lane & 15;
  int half = lane >> 4;
  int Mc = (M < 9) ? M : 0;          // keep addresses valid for padding lanes
  float rmask = (M < 9) ? 1.f : 0.f; // zero rows 9..15 (no EXEC manipulation)
  int dy = Mc / 3, dx = Mc % 3;

  v8f accLL = {}; v8f accPP = {}; v8f accLP = {};
  float sumL = 0.f, sumP = 0.f;

  const int STEPS = NPAIR / 4;             // 7056
  const int STRIDE = 8 * GSPLIT;           // 32 wave-streams per slot
  for (int s = part * 8 + wave; s < STEPS; s += STRIDE) {
    int n0 = 4 * s + 2 * half;
    v2f aL, aP;
#pragma unroll
    for (int j = 0; j < 2; ++j) {
      int n = n0 + j;
      int py = n / NWIN;
      int px = n - py * NWIN;
      float lv = la[(py + dy) * NPOOL + (px + dx)] * rmask;
      float pv = pr[(py + dy) * NPOOL + (px + dx)] * rmask;
      aL[j] = lv; aP[j] = pv;
      sumL += lv; sumP += pv;
    }
    accLL = __builtin_amdgcn_wmma_f32_16x16x4_f32(false, aL, false, aL, (short)0, accLL, false, false);
    accPP = __builtin_amdgcn_wmma_f32_16x16x4_f32(false, aP, false, aP, (short)0, accPP, false, false);
    accLP = __builtin_amdgcn_wmma_f32_16x16x4_f32(false, aL, false, aP, (short)0, accLP, false, false);
  }

  // D layout: VGPR v holds D[v + 8*half][lane%16]
  int col = M;
#pragma unroll
  for (int v = 0; v < 8; ++v) {
    int row = v + 8 * half;
    if (row < 9 && col < 9) {
      atomicAdd(&g[row * 9 + col],        accLL[v]);
      atomicAdd(&g[81 + row * 9 + col],   accPP[v]);
      atomicAdd(&g[162 + row * 9 + col],  accLP[v]);
    }
  }
  if (M < 9) {
    atomicAdd(&g[243 + M], sumL);
    atomicAdd(&g[252 + M], sumP);
  }
}

// ---------------------------------------------------------------------------
// Kernel 4: per-(b,c) 9x9 mean-correction, Cholesky solves, logdet.
// ---------------------------------------------------------------------------
__global__ void solve_kernel(const float* __restrict__ gram, float* __restrict__ scal) {
  int t = blockIdx.x * blockDim.x + threadIdx.x;
  if (t >= (int)N_SLOTS) return;
  int level = (t < BB * NF) ? 0 : ((t < BB * (NF + NM)) ? 1 : 2);
  const float* g = gram + (long long)t * GRAM_STRIDE;
  const float invN = 1.f / (float)NPAIR;

  float sl[9], sp[9];
  for (int d = 0; d < 9; ++d) { sl[d] = g[243 + d]; sp[d] = g[252 + d]; }
  float cll[9][9], cpp[9][9], clp[9][9];
  for (int d = 0; d < 9; ++d)
    for (int e = 0; e < 9; ++e) {
      cll[d][e] = g[d * 9 + e]       - sl[d] * sl[e] * invN;
      cpp[d][e] = g[81 + d * 9 + e]  - sp[d] * sp[e] * invN + ((d == e) ? ALPHA : 0.f);
      clp[d][e] = g[162 + d * 9 + e] - sl[d] * sp[e] * invN;
    }
  // Cholesky of pr_cov
  float Lc[9][9];
  for (int k = 0; k < 9; ++k) {
    float d = cpp[k][k];
    for (int j = 0; j < k; ++j) d -= Lc[k][j] * Lc[k][j];
    d = sqrtf(fmaxf(d, 1e-30f));
    Lc[k][k] = d;
    float inv = 1.f / d;
    for (int i = k + 1; i < 9; ++i) {
      float v = cpp[i][k];
      for (int j = 0; j < k; ++j) v -= Lc[i][j] * Lc[k][j];
      Lc[i][k] = v * inv;
    }
  }
  // V[i] = L^-1 * (row i of la_pr)ᵀ;  la_pr·pr_cov⁻¹·la_prᵀ = V·Vᵀ
  float V[9][9];
  for (int i = 0; i < 9; ++i)
    for (int k = 0; k < 9; ++k) {
      float v = clp[i][k];
      for (int j = 0; j < k; ++j) v -= Lc[k][j] * V[i][j];
      V[i][k] = v / Lc[k][k];
    }
  float A[9][9];
  for (int i = 0; i < 9; ++i)
    for (int j = 0; j < 9; ++j) {
      float dot = 0.f;
      for (int k = 0; k < 9; ++k) dot += V[i][k] * V[j][k];
      A[i][j] = cll[i][j] - dot + ((i == j) ? ALPHA : 0.f);
    }
  // Cholesky of appro, accumulate logdet = 2*sum(log(diag + 1e-8))
  float ld = 0.f;
  for (int k = 0; k < 9; ++k) {
    float d = A[k][k];
    for (int j = 0; j < k; ++j) d -= A[k][j] * A[k][j];
    d = sqrtf(fmaxf(d, 0.f));
    ld += 2.f * __logf(d + 1e-8f);
    A[k][k] = d;
    float inv = 1.f / fmaxf(d, 1e-30f);
    for (int i = k + 1; i < 9; ++i) {
      float v = A[i][k];
      for (int j = 0; j < k; ++j) v -= A[i][j] * A[k][j];
      A[i][k] = v * inv;
    }
  }
  atomicAdd(&scal[4 + level], ld);
}

// ---------------------------------------------------------------------------
// Kernel 5: assemble final scalar loss.
// ---------------------------------------------------------------------------
__global__ void final_kernel(const float* __restrict__ scal, float* __restrict__ out) {
  const float Cs[3] = {(float)NF, (float)NM, (float)NH};
  float total = 0.f;
#pragma unroll
  for (int l = 0; l < 3; ++l) {
    float bce = scal[l] / (scal[3] * Cs[l] + 1e-8f);
    float rmi = scal[4 + l] / (float)(BB * 9); // sum_c mean_b logdet / HALF_D
    total += 0.5f * bce + 0.5f * rmi;
  }
  out[0] = total;
}

// ---------------------------------------------------------------------------
extern "C" void kernel_launch(void* const* d_in, const int* in_sizes, int n_in,
                              void* d_out, int out_size, void* d_ws, size_t ws_size,
                              hipStream_t stream) {
  (void)in_sizes; (void)n_in; (void)out_size; (void)ws_size;
  // inputs: 0=step 1=embedding 2=cls_score_before (all unused by forward),
  //         3=cls_score (f32), 4=label (i32), 5/6=hierarchy maps (fixed constants)
  const float* cls = (const float*)d_in[3];
  const int* label = (const int*)d_in[4];
  float* ws = (float*)d_ws;
  float* out = (float*)d_out;

  // zero gram + scalar accumulators (graph-capturable memset node)
  hipMemsetAsync((char*)d_ws + OFF_GRAM * sizeof(float), 0,
                 (size_t)(N_SLOTS * GRAM_STRIDE + 8) * sizeof(float), stream);

  pool_bce_kernel<<<(BB * PP + 255) / 256, 256, 0, stream>>>(cls, label, ws);
  border_bce_kernel<<<(BB * (HWB - 510 * 510) + 255) / 256, 256, 0, stream>>>(
      cls, label, ws + OFF_SCAL);

  gram_kernel<<<BB * NF * GSPLIT, 256, 0, stream>>>(ws + OFF_LAF, ws + OFF_PRF, ws + OFF_GRAM);
  gram_kernel<<<BB * NM * GSPLIT, 256, 0, stream>>>(
      ws + OFF_LAM, ws + OFF_PRM, ws + OFF_GRAM + (long long)BB * NF * GRAM_STRIDE);
  gram_kernel<<<BB * NH * GSPLIT, 256, 0, stream>>>(
      ws + OFF_LAH, ws + OFF_PRH, ws + OFF_GRAM + (long long)BB * (NF + NM) * GRAM_STRIDE);

  solve_kernel<<<1, 128, 0, stream>>>(ws + OFF_GRAM, ws + OFF_SCAL);
  final_kernel<<<1, 1, 0, stream>>>(ws + OFF_SCAL, out);
}